// SpatioTemporalGCN_45724221833485
// MI455X (gfx1250) — compile-verified
//
#include <hip/hip_runtime.h>
#include <math.h>

typedef __attribute__((ext_vector_type(2))) float v2f;
typedef __attribute__((ext_vector_type(8))) float v8f;

#define WMMA_F32(a, b, c) \
  __builtin_amdgcn_wmma_f32_16x16x4_f32(false, (a), false, (b), (short)0, (c), false, false)

// ---------------------------------------------------------------------------
// Kernel 1: supports = softmax(relu(E @ E^T), axis=1)   E: (1024, 16)
// One block per row n; 256 threads, 4 columns per thread.
// ---------------------------------------------------------------------------
__global__ __launch_bounds__(256) void k_supports(const float* __restrict__ emb,
                                                  float* __restrict__ S) {
  const int n = blockIdx.x;
  const int t = threadIdx.x;
  __shared__ float e[16];
  __shared__ float red[256];
  if (t < 16) e[t] = emb[n * 16 + t];
  __syncthreads();
  float vals[4];
  float mx = -1e30f;
#pragma unroll
  for (int j = 0; j < 4; ++j) {
    const int m = t + j * 256;
    float acc = 0.f;
#pragma unroll
    for (int d = 0; d < 16; ++d) acc += e[d] * emb[m * 16 + d];
    acc = fmaxf(acc, 0.f);
    vals[j] = acc;
    mx = fmaxf(mx, acc);
  }
  red[t] = mx;
  __syncthreads();
  for (int s = 128; s > 0; s >>= 1) {
    if (t < s) red[t] = fmaxf(red[t], red[t + s]);
    __syncthreads();
  }
  const float rowmax = red[0];
  __syncthreads();
  float sum = 0.f;
#pragma unroll
  for (int j = 0; j < 4; ++j) {
    vals[j] = expf(vals[j] - rowmax);
    sum += vals[j];
  }
  red[t] = sum;
  __syncthreads();
  for (int s = 128; s > 0; s >>= 1) {
    if (t < s) red[t] += red[t + s];
    __syncthreads();
  }
  const float inv = 1.f / red[0];
#pragma unroll
  for (int j = 0; j < 4; ++j) S[(size_t)n * 1024 + t + j * 256] = vals[j] * inv;
}

// ---------------------------------------------------------------------------
// Kernel 2: sum_x[b, c] = sum_m x[b, m, c]
// ---------------------------------------------------------------------------
__global__ __launch_bounds__(256) void k_sumx(const float* __restrict__ x,
                                              float* __restrict__ sx) {
  const int b = blockIdx.x, t = threadIdx.x;
  const int c = t & 63, part = t >> 6;
  float acc = 0.f;
  for (int m = part * 256; m < part * 256 + 256; ++m)
    acc += x[((size_t)b * 1024 + m) * 64 + c];
  __shared__ float red[256];
  red[t] = acc;
  __syncthreads();
  if (t < 64) sx[b * 64 + t] = red[t] + red[t + 64] + red[t + 128] + red[t + 192];
}

// ---------------------------------------------------------------------------
// Kernel 3: Yout[b] = S @ Xin[b]   (S: 1024x1024, Xin: (64, 1024, 64))
// grid = (16 m-tiles, 64 batches); 256 threads = 8 waves, 2 C-tiles per wave.
// f32 WMMA 16x16x4; LDS-staged 64x64 A/B blocks (stride 68 to avoid conflicts).
// ---------------------------------------------------------------------------
__global__ __launch_bounds__(256) void k_spmm(const float* __restrict__ S,
                                              const float* __restrict__ Xin,
                                              float* __restrict__ Yout) {
  const int m0 = blockIdx.x * 64;
  const int b = blockIdx.y;
  __shared__ __attribute__((aligned(16))) float sA[64 * 68];
  __shared__ __attribute__((aligned(16))) float sB[64 * 68];
  const int t = threadIdx.x, l = t & 31, w = t >> 5;
  const int mt = w >> 1, nt0 = (w & 1) * 2;
  const int lm = l & 15, kp = (l >= 16) ? 2 : 0;
  const v8f vzero = {};
  v8f c0 = vzero, c1 = vzero;
  for (int k0 = 0; k0 < 1024; k0 += 64) {
    __syncthreads();
    for (int i4 = t; i4 < 1024; i4 += 256) {
      const int r = i4 >> 4, c4 = (i4 & 15) * 4;
      *(float4*)&sA[r * 68 + c4] = *(const float4*)&S[(size_t)(m0 + r) * 1024 + k0 + c4];
      *(float4*)&sB[r * 68 + c4] = *(const float4*)&Xin[((size_t)b * 1024 + k0 + r) * 64 + c4];
    }
    __syncthreads();
#pragma unroll
    for (int kk = 0; kk < 64; kk += 4) {
      v2f a, b0, b1;
      a.x = sA[(mt * 16 + lm) * 68 + kk + kp];
      a.y = sA[(mt * 16 + lm) * 68 + kk + kp + 1];
      b0.x = sB[(kk + kp) * 68 + nt0 * 16 + lm];
      b0.y = sB[(kk + kp + 1) * 68 + nt0 * 16 + lm];
      b1.x = sB[(kk + kp) * 68 + (nt0 + 1) * 16 + lm];
      b1.y = sB[(kk + kp + 1) * 68 + (nt0 + 1) * 16 + lm];
      c0 = WMMA_F32(a, b0, c0);
      c1 = WMMA_F32(a, b1, c1);
    }
  }
  const int mr = mt * 16 + ((l >= 16) ? 8 : 0);
#pragma unroll
  for (int r = 0; r < 8; ++r) {
    Yout[((size_t)b * 1024 + m0 + mr + r) * 64 + nt0 * 16 + lm] = c0[r];
    Yout[((size_t)b * 1024 + m0 + mr + r) * 64 + (nt0 + 1) * 16 + lm] = c1[r];
  }
}

// ---------------------------------------------------------------------------
// Shared 64x64x64 WMMA GEMM helpers (8 waves, 2 tiles/wave, accumulate).
// ---------------------------------------------------------------------------
__device__ __forceinline__ void gemm64_acc(const float* __restrict__ As, int lda,
                                           const float* __restrict__ Bs, int ldb,
                                           v8f& c0, v8f& c1, int t) {
  const int l = t & 31, w = t >> 5;
  const int mt = w >> 1, nt0 = (w & 1) * 2;
  const int lm = l & 15, kp = (l >= 16) ? 2 : 0;
#pragma unroll
  for (int kk = 0; kk < 64; kk += 4) {
    v2f a, b0, b1;
    a.x = As[(mt * 16 + lm) * lda + kk + kp];
    a.y = As[(mt * 16 + lm) * lda + kk + kp + 1];
    b0.x = Bs[(kk + kp) * ldb + nt0 * 16 + lm];
    b0.y = Bs[(kk + kp + 1) * ldb + nt0 * 16 + lm];
    b1.x = Bs[(kk + kp) * ldb + (nt0 + 1) * 16 + lm];
    b1.y = Bs[(kk + kp + 1) * ldb + (nt0 + 1) * 16 + lm];
    c0 = WMMA_F32(a, b0, c0);
    c1 = WMMA_F32(a, b1, c1);
  }
}

__device__ __forceinline__ void store_tiles(float* __restrict__ Out, int ldo,
                                            const v8f& c0, const v8f& c1, int t) {
  const int l = t & 31, w = t >> 5;
  const int mt = w >> 1, nt0 = (w & 1) * 2;
  const int lm = l & 15;
  const int mr = mt * 16 + ((l >= 16) ? 8 : 0);
#pragma unroll
  for (int r = 0; r < 8; ++r) {
    Out[(mr + r) * ldo + nt0 * 16 + lm] = c0[r];
    Out[(mr + r) * ldo + (nt0 + 1) * 16 + lm] = c1[r];
  }
}

// ---------------------------------------------------------------------------
// Kernel 4: per-node fused gconv / aconv / wconv + l2norm + bias + concat.
// One block (256 threads) per node n.
// ---------------------------------------------------------------------------
__global__ __launch_bounds__(256) void k_node(
    const float* __restrict__ x, const float* __restrict__ Y1, const float* __restrict__ Y2,
    const float* __restrict__ emb, const float* __restrict__ Wp, const float* __restrict__ Wpa,
    const float* __restrict__ Www, const float* __restrict__ biasp, const float* __restrict__ Tp,
    const float* __restrict__ adj, const float* __restrict__ sumx,
    const float* __restrict__ xw, float* __restrict__ out) {
  const int n = blockIdx.x;
  const int t = threadIdx.x;
  __shared__ __attribute__((aligned(16))) float sA[64 * 68];
  __shared__ __attribute__((aligned(16))) float sW[64 * 68];
  __shared__ __attribute__((aligned(16))) float sOut[64 * 64];
  __shared__ float sE[16];
  __shared__ float sBias[192];
  __shared__ float sDiag[3];
  __shared__ float sWW[65];
  __shared__ float sTp[12];
  if (t < 16) sE[t] = emb[n * 16 + t];
  if (t >= 32 && t < 35)
    sDiag[t - 32] = adj[(size_t)(t - 32) * 1024 * 1024 + (size_t)n * 1024 + n];
  if (t >= 64 && t < 76) sTp[t - 64] = Tp[t - 64];
  __syncthreads();
  if (t < 192) {
    float acc = 0.f;
#pragma unroll
    for (int d = 0; d < 16; ++d) acc += sE[d] * biasp[d * 192 + t];
    sBias[t] = acc;
  }

  // ---- gconv: (B=64) x (K=3*64) x (O=64), k-chunk c <-> {x, Y1, Y2} ----
  const v8f vzero = {};
  v8f c0 = vzero, c1 = vzero;
  for (int kc = 0; kc < 3; ++kc) {
    const float* src = (kc == 0) ? x : (kc == 1) ? Y1 : Y2;
    __syncthreads();
    for (int i4 = t; i4 < 1024; i4 += 256) {
      const int b = i4 >> 4, c4 = (i4 & 15) * 4;
      *(float4*)&sA[b * 68 + c4] = *(const float4*)&src[((size_t)b * 1024 + n) * 64 + c4];
    }
    for (int i4 = t; i4 < 1024; i4 += 256) {
      const int i = i4 >> 4, o4 = (i4 & 15) * 4;
      float4 acc = make_float4(0.f, 0.f, 0.f, 0.f);
#pragma unroll
      for (int d = 0; d < 16; ++d) {
        const float4 wv = *(const float4*)&Wp[(size_t)d * 12288 + kc * 4096 + i * 64 + o4];
        const float ed = sE[d];
        acc.x += ed * wv.x;
        acc.y += ed * wv.y;
        acc.z += ed * wv.z;
        acc.w += ed * wv.w;
      }
      *(float4*)&sW[i * 68 + o4] = acc;
    }
    __syncthreads();
    gemm64_acc(sA, 68, sW, 68, c0, c1, t);
  }
  __syncthreads();
  store_tiles(sOut, 64, c0, c1, t);
  __syncthreads();
  if (t < 64) {
    const int b = t;
    float ss = 0.f;
    for (int o = 0; o < 64; ++o) {
      const float v = sOut[b * 64 + o];
      ss += v * v;
    }
    const float inv = 1.f / fmaxf(sqrtf(ss), 1e-12f);
    for (int o = 0; o < 64; ++o)
      out[((size_t)b * 1024 + n) * 192 + o] = sOut[b * 64 + o] * inv + sBias[o];
  }

  // ---- aconv: x_a[b,n,k,c] = diag_k[n] * sum_x[b,c]  ->  (64x64)@(64x64) ----
  __syncthreads();
  for (int i4 = t; i4 < 1024; i4 += 256) {
    const int b = i4 >> 4, c4 = (i4 & 15) * 4;
    *(float4*)&sA[b * 68 + c4] = *(const float4*)&sumx[b * 64 + c4];
  }
  for (int i4 = t; i4 < 1024; i4 += 256) {
    const int i = i4 >> 4, o4 = (i4 & 15) * 4;
    float4 acc = make_float4(0.f, 0.f, 0.f, 0.f);
#pragma unroll
    for (int d = 0; d < 16; ++d) {
      const size_t base = (size_t)d * 12288 + i * 64 + o4;
      const float ed = sE[d];
      const float4 w0 = *(const float4*)&Wpa[base];
      const float4 w1 = *(const float4*)&Wpa[base + 4096];
      const float4 w2 = *(const float4*)&Wpa[base + 8192];
      acc.x += ed * (sDiag[0] * w0.x + sDiag[1] * w1.x + sDiag[2] * w2.x);
      acc.y += ed * (sDiag[0] * w0.y + sDiag[1] * w1.y + sDiag[2] * w2.y);
      acc.z += ed * (sDiag[0] * w0.z + sDiag[1] * w1.z + sDiag[2] * w2.z);
      acc.w += ed * (sDiag[0] * w0.w + sDiag[1] * w1.w + sDiag[2] * w2.w);
    }
    *(float4*)&sW[i * 68 + o4] = acc;
  }
  __syncthreads();
  c0 = vzero;
  c1 = vzero;
  gemm64_acc(sA, 68, sW, 68, c0, c1, t);
  store_tiles(sOut, 64, c0, c1, t);
  __syncthreads();
  if (t < 64) {
    const int b = t;
    float ss = 0.f;
    for (int o = 0; o < 64; ++o) {
      const float v = sOut[b * 64 + o];
      ss += v * v;
    }
    const float inv = 1.f / fmaxf(sqrtf(ss), 1e-12f);
    for (int o = 0; o < 64; ++o)
      out[((size_t)b * 1024 + n) * 192 + 64 + o] = sOut[b * 64 + o] * inv + sBias[64 + o];
  }

  // ---- wconv: rank-1  v[b,:] = s[b] * ww[:],  ||v|| = |s| * ||ww|| ----
  if (t < 64) {
    float acc = 0.f;
#pragma unroll
    for (int d = 0; d < 16; ++d) acc += sE[d] * Www[d * 64 + t];
    sWW[t] = acc;
  }
  __syncthreads();
  if (t == 0) {
    float ss = 0.f;
    for (int o = 0; o < 64; ++o) ss += sWW[o] * sWW[o];
    sWW[64] = sqrtf(ss);
  }
  __syncthreads();
  if (t < 64) {
    const int b = t;
    float s = 0.f;
#pragma unroll
    for (int tt = 0; tt < 12; ++tt)
      s += sTp[tt] * xw[((size_t)(b * 12 + tt)) * 1024 + n];
    const float inv = 1.f / fmaxf(fabsf(s) * sWW[64], 1e-12f);
    for (int o = 0; o < 64; ++o)
      out[((size_t)b * 1024 + n) * 192 + 128 + o] = s * sWW[o] * inv + sBias[128 + o];
  }
}

// ---------------------------------------------------------------------------
// Launch
// ---------------------------------------------------------------------------
extern "C" void kernel_launch(void* const* d_in, const int* in_sizes, int n_in,
                              void* d_out, int out_size, void* d_ws, size_t ws_size,
                              hipStream_t stream) {
  const float* x = (const float*)d_in[0];     // (64, 1024, 64)
  const float* xw = (const float*)d_in[1];    // (64, 12, 1024, 1)
  const float* emb = (const float*)d_in[2];   // (1024, 16)
  // d_in[3] = ZPI: unused by the reference computation
  const float* adj = (const float*)d_in[4];   // (3, 1024, 1024)
  const float* Wp = (const float*)d_in[5];    // (16, 3, 64, 64)
  const float* Wpa = (const float*)d_in[6];   // (16, 3, 64, 64)
  const float* Www = (const float*)d_in[7];   // (16, 1, 64)
  const float* bp = (const float*)d_in[8];    // (16, 192)
  const float* Tp = (const float*)d_in[9];    // (12,)
  float* out = (float*)d_out;                 // (64, 1024, 192)

  float* wsf = (float*)d_ws;
  float* supports = wsf;                      // 1024*1024      ( 4 MB)
  float* Y1 = supports + 1024 * 1024;         // 64*1024*64     (16 MB)
  float* Y2 = Y1 + 64 * 1024 * 64;            // 64*1024*64     (16 MB)
  float* sumx = Y2 + 64 * 1024 * 64;          // 64*64

  k_supports<<<1024, 256, 0, stream>>>(emb, supports);
  k_sumx<<<64, 256, 0, stream>>>(x, sumx);
  k_spmm<<<dim3(16, 64), 256, 0, stream>>>(supports, x, Y1);   // Y1 = S @ x
  k_spmm<<<dim3(16, 64), 256, 0, stream>>>(supports, Y1, Y2);  // Y2 = S @ Y1 = S^2 @ x
  k_node<<<1024, 256, 0, stream>>>(x, Y1, Y2, emb, Wp, Wpa, Www, bp, Tp, adj, sumx, xw, out);
}